// SpatialEncoding_65609920414078
// MI455X (gfx1250) — compile-verified
//
#include <hip/hip_runtime.h>

// ---------------------------------------------------------------------------
// SpatialEncoding: x = emb[feat]; 3x { acc = x + scatter_add(x[src]*w -> dst);
//                                      x = acc @ W.T + b }
// T=4, N=50000, E=600000, D=128.  N % 16 == 0 exactly (3125 row tiles).
// GEMM uses V_WMMA_F32_16X16X4_F32 (f32 in, f32 acc -> matches f32 reference).
// Scatter uses hardware global_atomic_add_f32 (L2 RMW), the bottleneck path.
// ---------------------------------------------------------------------------

typedef __attribute__((ext_vector_type(2))) float v2f;
typedef __attribute__((ext_vector_type(8))) float v8f;

#define DIM 128

// --- 1) Gather embedding rows; dual-store so ACC starts as x for iter 0. ---
__global__ void se_gather(const int* __restrict__ feats,
                          const float* __restrict__ emb,
                          float* __restrict__ X,
                          float* __restrict__ ACC,
                          int N) {
    int idx  = blockIdx.x * blockDim.x + threadIdx.x;   // N * 32 lanes
    int n    = idx >> 5;
    int lane = idx & 31;
    if (n >= N) return;
    const float4* src = (const float4*)(emb + (size_t)feats[n] * DIM);
    float4 v = src[lane];
    ((float4*)(X   + (size_t)n * DIM))[lane] = v;
    ((float4*)(ACC + (size_t)n * DIM))[lane] = v;
}

// --- 2) ACC = X (accumulator re-init between propagate iterations). --------
__global__ void se_copy(const float* __restrict__ X,
                        float* __restrict__ ACC,
                        int nVec4) {
    int idx = blockIdx.x * blockDim.x + threadIdx.x;
    if (idx < nVec4) ((float4*)ACC)[idx] = ((const float4*)X)[idx];
}

// --- 3) Edge scatter: ACC[dst] += X[src] * w.  One wave32 per edge, ---------
//        each lane handles 4 contiguous floats of the 128-wide row.
__global__ void se_scatter(const int* __restrict__ src,
                           const int* __restrict__ dst,
                           const float* __restrict__ wgt,
                           const float* __restrict__ X,
                           float* __restrict__ ACC,
                           int E) {
    int idx  = blockIdx.x * blockDim.x + threadIdx.x;   // E * 32 lanes
    int e    = idx >> 5;
    int lane = idx & 31;
    if (e >= E) return;
    int   s = src[e];
    int   d = dst[e];
    float w = wgt[e];
    float4 v = ((const float4*)(X + (size_t)s * DIM))[lane];
    float* out = ACC + (size_t)d * DIM + lane * 4;
    unsafeAtomicAdd(out + 0, v.x * w);   // -> global_atomic_add_f32 (L2 RMW)
    unsafeAtomicAdd(out + 1, v.y * w);
    unsafeAtomicAdd(out + 2, v.z * w);
    unsafeAtomicAdd(out + 3, v.w * w);
}

// --- 4) WMMA GEMM: Xout[m, n] = sum_k Y[m,k] * W[n,k] + b[n]  (i.e. Y@W.T+b)
//        Block = 256 threads = 8 waves; wave w owns the 16x16 tile at
//        (rowBase = 16*blockIdx.x, colBase = 16*w).  k-loop: 32 WMMAs of K=4.
//
//        A (16x4 f32) lane layout: lanes 0-15 rows M=0..15, VGPR0/1 = K,K+1;
//        lanes 16-31 same rows, VGPR0/1 = K+2,K+3  -> float2 load at k+2*half.
//        B (4x16) mirrors the K-split across lane halves; B[k'][n] = W[n][k'],
//        so lane (half,l) loads float2 from W row (colBase+l) at k+2*half.
//        C/D (16x16 f32): VGPR v -> row v + 8*half, col = l.
__global__ void se_gemm_wmma(const float* __restrict__ Y,
                             const float* __restrict__ W,
                             const float* __restrict__ b,
                             float* __restrict__ Xout,
                             int N) {
    const int wave = threadIdx.x >> 5;          // 0..7 -> column tile
    const int lane = threadIdx.x & 31;
    const int half = lane >> 4;                 // 0 or 1 (K split)
    const int l    = lane & 15;                 // row (A) / col (B,C)
    const int rowBase = blockIdx.x * 16;
    const int colBase = wave * 16;

    const float* yrow = Y + (size_t)(rowBase + l) * DIM + 2 * half;
    const float* wrow = W + (size_t)(colBase + l) * DIM + 2 * half;

    v8f c = {};
#pragma unroll
    for (int k = 0; k < DIM; k += 4) {
        v2f a  = *(const v2f*)(yrow + k);
        v2f bb = *(const v2f*)(wrow + k);
        // D = A x B + C ; f32 path, round-to-nearest-even, matches reference.
        c = __builtin_amdgcn_wmma_f32_16x16x4_f32(
                /*neg_a=*/false, a, /*neg_b=*/false, bb,
                /*c_mod=*/(short)0, c, /*reuse_a=*/false, /*reuse_b=*/false);
    }

    const float bias = b[colBase + l];
#pragma unroll
    for (int v = 0; v < 8; ++v) {
        Xout[(size_t)(rowBase + v + 8 * half) * DIM + (colBase + l)] = c[v] + bias;
    }
}

extern "C" void kernel_launch(void* const* d_in, const int* in_sizes, int n_in,
                              void* d_out, int out_size, void* d_ws, size_t ws_size,
                              hipStream_t stream) {
    const int T = 4;
    const int N = in_sizes[0] / T;           // 50000
    const int E = in_sizes[2] / T;           // 600000

    const int*   node_features = (const int*)d_in[0];    // [T, N]
    const int*   edge_index    = (const int*)d_in[1];    // [T, 2, E]
    const float* edge_weights  = (const float*)d_in[2];  // [T, E]
    const float* emb_table     = (const float*)d_in[3];  // [VOCAB, 128]
    const float* Wm            = (const float*)d_in[4];  // [128, 128]
    const float* bv            = (const float*)d_in[5];  // [128]
    float*       out           = (float*)d_out;          // [T, N, 128]

    float* X   = (float*)d_ws;                 // [N, 128]
    float* ACC = X + (size_t)N * DIM;          // [N, 128]

    const int threads = 256;
    const int gatherBlocks  = (N * 32 + threads - 1) / threads;
    const int copyBlocks    = (N * 32 + threads - 1) / threads;   // N*128/4 vec4s
    const int scatterBlocks = (int)(((size_t)E * 32 + threads - 1) / threads);
    const int gemmBlocks    = N / 16;          // N % 16 == 0

    for (int t = 0; t < T; ++t) {
        const int*   feats_t = node_features + (size_t)t * N;
        const int*   src_t   = edge_index + (size_t)t * 2 * E;
        const int*   dst_t   = src_t + E;
        const float* w_t     = edge_weights + (size_t)t * E;

        se_gather<<<gatherBlocks, threads, 0, stream>>>(feats_t, emb_table, X, ACC, N);

        for (int it = 0; it < 3; ++it) {
            if (it > 0)
                se_copy<<<copyBlocks, threads, 0, stream>>>(X, ACC, N * 32);

            se_scatter<<<scatterBlocks, threads, 0, stream>>>(src_t, dst_t, w_t, X, ACC, E);

            float* dstBuf = (it == 2) ? (out + (size_t)t * N * DIM) : X;
            se_gemm_wmma<<<gemmBlocks, threads, 0, stream>>>(ACC, Wm, bv, dstBuf, N);
        }
    }
}